// NutriGraphNet_11098195493029
// MI455X (gfx1250) — compile-verified
//
#include <hip/hip_runtime.h>
#include <math.h>

// ---------------------------------------------------------------------------
// NutriGraphNet for MI455X (gfx1250, wave32).
// Dense 64-wide transforms + gathered decoder GEMM use V_WMMA_F32_16X16X4_F32
// (exact fp32 accumulate -> numerically matches the fp32 reference).
// A-fragments: clamped row pointers + b64 vector loads, fully preloaded ahead
// of the WMMA chain (no exec-mask branches, no per-step loadcnt waits).
// B-fragments: W swizzled in LDS so each (k,k+1) pair is one ds_load_b64.
// Edge softmax/aggregation is HBM-bandwidth bound -> atomic 3-pass scheme.
// ---------------------------------------------------------------------------

#define NU      100000
#define NF      30000
#define E_RATES 1000000
#define E_REV   1000000
#define E_HLT   500000
#define E_LBL   250000
#define EPSBN   1e-5f

typedef float v2f __attribute__((ext_vector_type(2)));
typedef float v8f __attribute__((ext_vector_type(8)));

static __host__ int divUp(int a, int b) { return (a + b - 1) / b; }

// swizzled LDS index for W[k][n] so that (k, k+1) pairs are contiguous:
// k = 4q + 2kh + j  ->  Ws[((q*2 + kh)*64 + n)*2 + j]
static __device__ __forceinline__ int wswz(int k, int n) {
  return ((((k >> 2) << 1) + ((k >> 1) & 1)) * 64 + n) * 2 + (k & 1);
}

// float atomic max via integer punning (works for all sign combinations,
// memory must be initialized to -inf).
static __device__ __forceinline__ void atomicMaxF(float* addr, float v) {
  if (v >= 0.0f) atomicMax((int*)addr, __float_as_int(v));
  else           atomicMin((unsigned int*)addr, (unsigned int)__float_as_int(v));
}

__global__ void k_fill(float* __restrict__ p, float v, int n) {
  int i = blockIdx.x * blockDim.x + threadIdx.x;
  if (i < n) p[i] = v;
}

// ---------------------------------------------------------------------------
// H[N,64] = X[N,64] @ W[64,64]; one wave32 per 16 rows, 8 waves/block.
// ---------------------------------------------------------------------------
__global__ __launch_bounds__(256) void k_gemm64(const float* __restrict__ X,
                                                const float* __restrict__ W,
                                                float* __restrict__ H, int N) {
  __shared__ float Ws[64 * 64];
  for (int i = threadIdx.x; i < 64 * 64; i += 256)
    Ws[wswz(i >> 6, i & 63)] = W[i];
  __syncthreads();
  const int lane = threadIdx.x & 31;
  const int wave = threadIdx.x >> 5;
  const int row0 = (blockIdx.x * 8 + wave) * 16;
  const int m    = lane & 15;
  const int kh   = lane >> 4;                 // 0/1 -> K pair 2*kh within group
  int r  = row0 + m;
  int rc = r < N ? r : (N - 1);               // clamp: loads never diverge
  const float* xr = X + (size_t)rc * 64 + 2 * kh;
  v2f a[16];
#pragma unroll
  for (int q = 0; q < 16; q++) a[q] = *(const v2f*)(xr + 4 * q);
  v8f acc0 = {}, acc1 = {}, acc2 = {}, acc3 = {};
#pragma unroll
  for (int q = 0; q < 16; q++) {
    const v2f* wq = (const v2f*)(Ws + (((q << 1) + kh) * 64) * 2);
    v2f b0 = wq[m];
    v2f b1 = wq[16 + m];
    v2f b2 = wq[32 + m];
    v2f b3 = wq[48 + m];
    acc0 = __builtin_amdgcn_wmma_f32_16x16x4_f32(false, a[q], false, b0, (short)0, acc0, false, false);
    acc1 = __builtin_amdgcn_wmma_f32_16x16x4_f32(false, a[q], false, b1, (short)0, acc1, false, false);
    acc2 = __builtin_amdgcn_wmma_f32_16x16x4_f32(false, a[q], false, b2, (short)0, acc2, false, false);
    acc3 = __builtin_amdgcn_wmma_f32_16x16x4_f32(false, a[q], false, b3, (short)0, acc3, false, false);
  }
  const int n = lane & 15;
#pragma unroll
  for (int i = 0; i < 8; i++) {
    int rr = row0 + i + (lane >> 4) * 8;
    if (rr < N) {
      float* hr = H + (size_t)rr * 64;
      hr[n]      = acc0[i];
      hr[16 + n] = acc1[i];
      hr[32 + n] = acc2[i];
      hr[48 + n] = acc3[i];
    }
  }
}

// out[n] = H[n,:64] . v[:64]  (one wave per node)
__global__ __launch_bounds__(256) void k_rowdot(const float* __restrict__ H,
                                                const float* __restrict__ v,
                                                float* __restrict__ out, int N) {
  int lane = threadIdx.x & 31;
  int node = blockIdx.x * 8 + (threadIdx.x >> 5);
  if (node >= N) return;
  const float* h = H + (size_t)node * 64;
  float s = h[lane] * v[lane] + h[32 + lane] * v[32 + lane];
  for (int o = 16; o; o >>= 1) s += __shfl_down(s, o, 32);
  if (lane == 0) out[node] = s;
}

// wtilde[k] = sum_n W[k,n] * att[n]    (single 64-thread block)
__global__ void k_mv64(const float* __restrict__ W, const float* __restrict__ att,
                       float* __restrict__ out) {
  int k = threadIdx.x;
  float s = 0.0f;
  for (int n = 0; n < 64; n++) s += W[k * 64 + n] * att[n];
  out[k] = s;
}

__global__ void k_edge_logit_max(const int* __restrict__ src, const int* __restrict__ dst,
                                 const float* __restrict__ as_, const float* __restrict__ ad_,
                                 float* __restrict__ ea, float* __restrict__ m_, int E) {
  int e = blockIdx.x * blockDim.x + threadIdx.x;
  if (e >= E) return;
  float a = as_[src[e]] + ad_[dst[e]];
  a = a > 0.0f ? a : 0.2f * a;              // leaky_relu(0.2)
  ea[e] = a;
  atomicMaxF(&m_[dst[e]], a);
}

__global__ void k_edge_expsum(const int* __restrict__ dst, float* __restrict__ ea,
                              const float* __restrict__ m_, float* __restrict__ s_, int E) {
  int e = blockIdx.x * blockDim.x + threadIdx.x;
  if (e >= E) return;
  int d = dst[e];
  float ex = __expf(ea[e] - m_[d]);
  ea[e] = ex;
  atomicAdd(&s_[d], ex);
}

// out[dst,:] += (ex/s[dst]) * hs[src,:]   (one wave per edge; 2 features/lane)
__global__ __launch_bounds__(256) void k_edge_scatter(const int* __restrict__ src,
                                                      const int* __restrict__ dst,
                                                      const float* __restrict__ ea,
                                                      const float* __restrict__ s_,
                                                      const float* __restrict__ hs,
                                                      float* __restrict__ out, int E) {
  int lane = threadIdx.x & 31;
  int e = blockIdx.x * 8 + (threadIdx.x >> 5);
  if (e >= E) return;
  int sn = src[e], d = dst[e];
  float alpha = ea[e] / s_[d];
  const float* h = hs + (size_t)sn * 64;
  float* o = out + (size_t)d * 64;
  atomicAdd(&o[lane], alpha * h[lane]);
  atomicAdd(&o[32 + lane], alpha * h[32 + lane]);
}

// column sums + sums of squares for 64-wide matrices (training-mode BN stats)
__global__ __launch_bounds__(256) void k_bn_stats(const float* __restrict__ X,
                                                  float* __restrict__ sums, int N) {
  __shared__ float s1[64], s2[64];
  if (threadIdx.x < 64) { s1[threadIdx.x] = 0.0f; s2[threadIdx.x] = 0.0f; }
  __syncthreads();
  int c = threadIdx.x & 63;
  float a = 0.0f, b = 0.0f;
  for (int r = blockIdx.x * 4 + (threadIdx.x >> 6); r < N; r += gridDim.x * 4) {
    float x = X[(size_t)r * 64 + c];
    a += x; b += x * x;
  }
  atomicAdd(&s1[c], a);
  atomicAdd(&s2[c], b);
  __syncthreads();
  if (threadIdx.x < 64) {
    atomicAdd(&sums[threadIdx.x], s1[threadIdx.x]);
    atomicAdd(&sums[64 + threadIdx.x], s2[threadIdx.x]);
  }
}

// x = elu( (x - mu) * rsqrt(var + eps) * g + b )   in place
__global__ void k_bn_elu(float* __restrict__ X, const float* __restrict__ sums,
                         const float* __restrict__ g, const float* __restrict__ b, int N) {
  int i = blockIdx.x * blockDim.x + threadIdx.x;
  if (i >= N * 64) return;
  int c = i & 63;
  float inv = 1.0f / (float)N;
  float mu = sums[c] * inv;
  float var = sums[64 + c] * inv - mu * mu;  // biased variance (ddof=0)
  float y = (X[i] - mu) * rsqrtf(var + EPSBN) * g[c] + b[c];
  X[i] = y > 0.0f ? y : (__expf(y) - 1.0f);
}

// pref = sigmoid( relu(xu @ W1 + b1) @ W2 + b2 )  -- one wave/node, lane = hidden unit
__global__ __launch_bounds__(256) void k_hp(const float* __restrict__ xu,
                                            const float* __restrict__ W1,
                                            const float* __restrict__ b1,
                                            const float* __restrict__ W2,
                                            const float* __restrict__ b2,
                                            float* __restrict__ pref, int N) {
  __shared__ float Wl[64 * 32];
  for (int i = threadIdx.x; i < 64 * 32; i += 256) Wl[i] = W1[i];
  __syncthreads();
  int lane = threadIdx.x & 31;
  int node = blockIdx.x * 8 + (threadIdx.x >> 5);
  if (node >= N) return;
  const float* x = xu + (size_t)node * 64;
  float h = b1[lane];
#pragma unroll
  for (int k = 0; k < 64; k++) h += x[k] * Wl[k * 32 + lane];
  h = h > 0.0f ? h : 0.0f;
  float s = h * W2[lane];
  for (int o = 16; o; o >>= 1) s += __shfl_down(s, o, 32);
  if (lane == 0) pref[node] = 1.0f / (1.0f + __expf(-(s + b2[0])));
}

__global__ void k_health(const float* __restrict__ pref, const float* __restrict__ scr,
                         const int* __restrict__ hu, const int* __restrict__ hf,
                         float* __restrict__ fupd, int E) {
  int e = blockIdx.x * blockDim.x + threadIdx.x;
  if (e >= E) return;
  atomicAdd(&fupd[hf[e]], pref[hu[e]] * scr[e]);
}

__global__ void k_xf_add(float* __restrict__ xf, const float* __restrict__ fupd, int N) {
  int i = blockIdx.x * blockDim.x + threadIdx.x;
  if (i >= N * 64) return;
  xf[i] += 0.1f * fupd[i >> 6];
}

// Z[E,64] = concat(xu[ls], xf[ld]) @ W1[128,64]  (gathered-A WMMA GEMM)
__global__ __launch_bounds__(256) void k_dec_gemm(const float* __restrict__ xu,
                                                  const float* __restrict__ xf,
                                                  const int* __restrict__ ls,
                                                  const int* __restrict__ ld,
                                                  const float* __restrict__ W1,
                                                  float* __restrict__ Z, int E) {
  __shared__ float Ws[128 * 64];
  for (int i = threadIdx.x; i < 128 * 64; i += 256)
    Ws[wswz(i >> 6, i & 63)] = W1[i];
  __syncthreads();
  const int lane = threadIdx.x & 31;
  const int wave = threadIdx.x >> 5;
  const int row0 = (blockIdx.x * 8 + wave) * 16;
  const int m    = lane & 15;
  const int kh   = lane >> 4;
  int r  = row0 + m;
  int rc = r < E ? r : (E - 1);                        // clamp
  const float* ru = xu + (size_t)ls[rc] * 64 + 2 * kh;
  const float* rf = xf + (size_t)ld[rc] * 64 + 2 * kh;
  v8f acc0 = {}, acc1 = {}, acc2 = {}, acc3 = {};
  v2f a[16];
  // --- K = 0..63 (user half) ---
#pragma unroll
  for (int q = 0; q < 16; q++) a[q] = *(const v2f*)(ru + 4 * q);
#pragma unroll
  for (int q = 0; q < 16; q++) {
    const v2f* wq = (const v2f*)(Ws + (((q << 1) + kh) * 64) * 2);
    v2f b0 = wq[m], b1 = wq[16 + m], b2 = wq[32 + m], b3 = wq[48 + m];
    acc0 = __builtin_amdgcn_wmma_f32_16x16x4_f32(false, a[q], false, b0, (short)0, acc0, false, false);
    acc1 = __builtin_amdgcn_wmma_f32_16x16x4_f32(false, a[q], false, b1, (short)0, acc1, false, false);
    acc2 = __builtin_amdgcn_wmma_f32_16x16x4_f32(false, a[q], false, b2, (short)0, acc2, false, false);
    acc3 = __builtin_amdgcn_wmma_f32_16x16x4_f32(false, a[q], false, b3, (short)0, acc3, false, false);
  }
  // --- K = 64..127 (food half) ---
#pragma unroll
  for (int q = 0; q < 16; q++) a[q] = *(const v2f*)(rf + 4 * q);
#pragma unroll
  for (int q = 0; q < 16; q++) {
    const int qq = q + 16;
    const v2f* wq = (const v2f*)(Ws + (((qq << 1) + kh) * 64) * 2);
    v2f b0 = wq[m], b1 = wq[16 + m], b2 = wq[32 + m], b3 = wq[48 + m];
    acc0 = __builtin_amdgcn_wmma_f32_16x16x4_f32(false, a[q], false, b0, (short)0, acc0, false, false);
    acc1 = __builtin_amdgcn_wmma_f32_16x16x4_f32(false, a[q], false, b1, (short)0, acc1, false, false);
    acc2 = __builtin_amdgcn_wmma_f32_16x16x4_f32(false, a[q], false, b2, (short)0, acc2, false, false);
    acc3 = __builtin_amdgcn_wmma_f32_16x16x4_f32(false, a[q], false, b3, (short)0, acc3, false, false);
  }
  const int n = lane & 15;
#pragma unroll
  for (int i = 0; i < 8; i++) {
    int rr = row0 + i + (lane >> 4) * 8;
    if (rr < E) {
      float* zr = Z + (size_t)rr * 64;
      zr[n]      = acc0[i];
      zr[16 + n] = acc1[i];
      zr[32 + n] = acc2[i];
      zr[48 + n] = acc3[i];
    }
  }
}

// out = sigmoid( relu(BN(Z)) @ W2 + b2 )   (one wave per edge)
__global__ __launch_bounds__(256) void k_dec_final(const float* __restrict__ Z,
                                                   const float* __restrict__ sums,
                                                   const float* __restrict__ g,
                                                   const float* __restrict__ b,
                                                   const float* __restrict__ W2,
                                                   const float* __restrict__ b2,
                                                   float* __restrict__ out, int E) {
  int lane = threadIdx.x & 31;
  int e = blockIdx.x * 8 + (threadIdx.x >> 5);
  if (e >= E) return;
  float inv = 1.0f / (float)E;
  float s = 0.0f;
#pragma unroll
  for (int h = 0; h < 2; h++) {
    int c = lane + 32 * h;
    float mu = sums[c] * inv;
    float var = sums[64 + c] * inv - mu * mu;
    float y = (Z[(size_t)e * 64 + c] - mu) * rsqrtf(var + EPSBN) * g[c] + b[c];
    y = y > 0.0f ? y : 0.0f;
    s += y * W2[c];
  }
  for (int o = 16; o; o >>= 1) s += __shfl_down(s, o, 32);
  if (lane == 0) out[e] = 1.0f / (1.0f + __expf(-(s + b2[0])));
}

// ---------------------------------------------------------------------------
// Host orchestration
// ---------------------------------------------------------------------------
struct GatBufs {
  float *hs, *as_, *ad_, *m_, *s_, *ea, *wt;
};

static void launch_gat(hipStream_t st, const float* xsrc, const float* xdst,
                       const float* Wsrc, const float* atts,
                       const float* Wdst, const float* attd,
                       const int* src, const int* dst, int E, int Ns, int Nd,
                       float* msg_out, const GatBufs& b) {
  k_gemm64<<<divUp(Ns, 128), 256, 0, st>>>(xsrc, Wsrc, b.hs, Ns);
  k_rowdot<<<divUp(Ns, 8), 256, 0, st>>>(b.hs, atts, b.as_, Ns);
  k_mv64<<<1, 64, 0, st>>>(Wdst, attd, b.wt);          // ad = x_dst . (W_dst @ att_dst)
  k_rowdot<<<divUp(Nd, 8), 256, 0, st>>>(xdst, b.wt, b.ad_, Nd);
  k_fill<<<divUp(Nd, 256), 256, 0, st>>>(b.m_, -INFINITY, Nd);
  k_fill<<<divUp(Nd, 256), 256, 0, st>>>(b.s_, 0.0f, Nd);
  k_edge_logit_max<<<divUp(E, 256), 256, 0, st>>>(src, dst, b.as_, b.ad_, b.ea, b.m_, E);
  k_edge_expsum<<<divUp(E, 256), 256, 0, st>>>(dst, b.ea, b.m_, b.s_, E);
  k_edge_scatter<<<divUp(E, 8), 256, 0, st>>>(src, dst, b.ea, b.s_, b.hs, msg_out, E);
  // GAT bias skipped: constant per-feature shift cancels exactly in training-mode BN.
}

extern "C" void kernel_launch(void* const* d_in, const int* in_sizes, int n_in,
                              void* d_out, int out_size, void* d_ws, size_t ws_size,
                              hipStream_t stream) {
  // --- input map: top-level dict in insertion order; 'params' pytree
  //     flattened with sorted keys (JAX convention) -> 48 leaves at [2..49].
  const float* xu0 = (const float*)d_in[0];
  const float* xf0 = (const float*)d_in[1];
  auto F = [&](int i) { return (const float*)d_in[i]; };
  auto I = [&](int i) { return (const int*)d_in[i]; };
  // bn0: food_b, food_g, user_b, user_g  (2..5); bn1: (6..9)
  // convL.{health,rates,rev}.{W_dst,W_src,att_dst,att_src,bias}: 10..24, 25..39
  // dec.{W1,W2,b1,b2,bn_b,bn_g}: 40..45 ; hp.{W1,W2,b1,b2}: 46..49
  const int C0 = 10, C1 = 25;
  const float* hscore = F(50);
  const int *rat_s = I(51), *rat_d = I(52), *rev_s = I(53), *rev_d = I(54);
  const int *hlt_s = I(55), *hlt_d = I(56), *he_u = I(57), *he_f = I(58);
  const int *lbl_s = I(59), *lbl_d = I(60);

  // --- workspace bump allocator (floats)
  float* ws = (float*)d_ws;
  size_t off = 0;
  auto alloc = [&](size_t n) { float* p = ws + off; off += (n + 63) & ~(size_t)63; return p; };
  float* xu_a = alloc((size_t)NU * 64);
  float* xf_a = alloc((size_t)NF * 64);
  float* xu_b = alloc((size_t)NU * 64);
  float* xf_b = alloc((size_t)NF * 64);
  GatBufs gb;
  gb.hs  = alloc((size_t)NU * 64);
  gb.as_ = alloc(NU);
  gb.ad_ = alloc(NU);
  gb.m_  = alloc(NU);
  gb.s_  = alloc(NU);
  gb.ea  = alloc(E_RATES);
  gb.wt  = alloc(64);
  float* fupd  = alloc(NF);
  float* bnsum = alloc(128);
  float* Z     = alloc((size_t)E_LBL * 64);
  (void)ws_size; (void)in_sizes; (void)n_in; (void)out_size;

  float* out_lbl  = (float*)d_out;          // [E_LBL]
  float* out_pref = (float*)d_out + E_LBL;  // [NU]

  const float* xu = xu0; const float* xf = xf0;
  float* nxu[2] = { xu_a, xu_b };
  float* nxf[2] = { xf_a, xf_b };
  for (int l = 0; l < 2; l++) {
    int cv = (l == 0) ? C0 : C1;
    int bn = (l == 0) ? 2 : 6;
    float* fmsg = nxf[l];
    float* umsg = nxu[l];
    k_fill<<<divUp(NF * 64, 256), 256, 0, stream>>>(fmsg, 0.0f, NF * 64);
    k_fill<<<divUp(NU * 64, 256), 256, 0, stream>>>(umsg, 0.0f, NU * 64);
    // health + rates:  users -> foods (accumulate into fmsg; HeteroConv sum)
    launch_gat(stream, xu, xf, F(cv + 1), F(cv + 3), F(cv + 0), F(cv + 2),
               hlt_s, hlt_d, E_HLT, NU, NF, fmsg, gb);
    launch_gat(stream, xu, xf, F(cv + 6), F(cv + 8), F(cv + 5), F(cv + 7),
               rat_s, rat_d, E_RATES, NU, NF, fmsg, gb);
    // rev: foods -> users
    launch_gat(stream, xf, xu, F(cv + 11), F(cv + 13), F(cv + 10), F(cv + 12),
               rev_s, rev_d, E_REV, NF, NU, umsg, gb);
    // BN (training stats) + ELU, in place
    k_fill<<<1, 128, 0, stream>>>(bnsum, 0.0f, 128);
    k_bn_stats<<<512, 256, 0, stream>>>(umsg, bnsum, NU);
    k_bn_elu<<<divUp(NU * 64, 256), 256, 0, stream>>>(umsg, bnsum, F(bn + 3), F(bn + 2), NU);
    k_fill<<<1, 128, 0, stream>>>(bnsum, 0.0f, 128);
    k_bn_stats<<<512, 256, 0, stream>>>(fmsg, bnsum, NF);
    k_bn_elu<<<divUp(NF * 64, 256), 256, 0, stream>>>(fmsg, bnsum, F(bn + 1), F(bn + 0), NF);
    xu = umsg; xf = fmsg;
  }

  // health-preference MLP (writes pref directly into d_out tail)
  k_hp<<<divUp(NU, 8), 256, 0, stream>>>(xu, F(46), F(48), F(47), F(49), out_pref, NU);
  // xf += 0.1 * segment_sum(pref[he_user] * scores, he_food)
  k_fill<<<divUp(NF, 256), 256, 0, stream>>>(fupd, 0.0f, NF);
  k_health<<<divUp(E_HLT, 256), 256, 0, stream>>>(out_pref, hscore, he_u, he_f, fupd, E_HLT);
  k_xf_add<<<divUp(NF * 64, 256), 256, 0, stream>>>(nxf[1], fupd, NF);

  // edge decoder: gathered WMMA GEMM -> BN stats -> fused BN+relu+dot+sigmoid
  // (dec b1 cancels in training-mode BN and is skipped)
  k_dec_gemm<<<divUp(E_LBL, 128), 256, 0, stream>>>(nxu[1], nxf[1], lbl_s, lbl_d, F(40), Z, E_LBL);
  k_fill<<<1, 128, 0, stream>>>(bnsum, 0.0f, 128);
  k_bn_stats<<<512, 256, 0, stream>>>(Z, bnsum, E_LBL);
  k_dec_final<<<divUp(E_LBL, 8), 256, 0, stream>>>(Z, bnsum, F(45), F(44), F(41), F(43),
                                                   out_lbl, E_LBL);
}